// HungarianMatcher_72584947302562
// MI455X (gfx1250) — compile-verified
//
#include <hip/hip_runtime.h>
#include <math.h>

// Problem constants (from reference: bs=16, nq=512, ncls=2, T=512)
#define NROWS 8192   // bs*nq
#define NT    512    // targets
#define KD    53     // keypoint dim: [0:2)=C, [2:36)=Z (34), [36:53)=V (17)
#define PADW  24     // padded fragment row: [0..16]=V, [17..19]=0, [20..21]=C, [22..23]=0

typedef float v2f __attribute__((ext_vector_type(2)));
typedef float v8f __attribute__((ext_vector_type(8)));

// Single-instruction v_sqrt_f32 (inputs are clamped >= 1e-12, so the precise
// sequence's denormal-range fixup is unnecessary).
__device__ __forceinline__ float fast_sqrtf(float x) {
  return __builtin_amdgcn_sqrtf(x);
}

// ---------------------------------------------------------------------------
// Kernel 1: per-row precompute.
//   pred rows n: negP = -softmax(logits[n]); nVp/nCp squared norms; padded row
//   tgt  rows t: nVg/nCg squared norms; padded row
// Padded rows make every WMMA fragment load an unconditional aligned b64 —
// the zero pad replaces per-element predication (no exec divergence).
// ---------------------------------------------------------------------------
__global__ __launch_bounds__(256) void prep_kernel(
    const float* __restrict__ logits, const float* __restrict__ pkp,
    const float* __restrict__ tkp,
    float* __restrict__ negP, float* __restrict__ nVp, float* __restrict__ nCp,
    float* __restrict__ nVg, float* __restrict__ nCg,
    float* __restrict__ padP, float* __restrict__ padT) {
  int i = blockIdx.x * blockDim.x + threadIdx.x;
  if (i < NROWS) {
    float l0 = logits[2 * i], l1 = logits[2 * i + 1];
    float m = fmaxf(l0, l1);
    float e0 = expf(l0 - m), e1 = expf(l1 - m);
    float inv = 1.0f / (e0 + e1);
    negP[2 * i]     = -e0 * inv;
    negP[2 * i + 1] = -e1 * inv;
    const float* r = pkp + i * KD;
    float sv = 0.f;
#pragma unroll
    for (int k = 36; k < 53; ++k) sv = fmaf(r[k], r[k], sv);
    nVp[i] = sv;
    nCp[i] = fmaf(r[0], r[0], r[1] * r[1]);
    float* p = padP + i * PADW;
#pragma unroll
    for (int k = 0; k < 17; ++k) p[k] = r[36 + k];
    p[17] = 0.f; p[18] = 0.f; p[19] = 0.f;
    p[20] = r[0]; p[21] = r[1]; p[22] = 0.f; p[23] = 0.f;
  } else if (i < NROWS + NT) {
    int t = i - NROWS;
    const float* r = tkp + t * KD;
    float sv = 0.f;
#pragma unroll
    for (int k = 36; k < 53; ++k) sv = fmaf(r[k], r[k], sv);
    nVg[t] = sv;
    nCg[t] = fmaf(r[0], r[0], r[1] * r[1]);
    float* p = padT + t * PADW;
#pragma unroll
    for (int k = 0; k < 17; ++k) p[k] = r[36 + k];
    p[17] = 0.f; p[18] = 0.f; p[19] = 0.f;
    p[20] = r[0]; p[21] = r[1]; p[22] = 0.f; p[23] = 0.f;
  }
}

// ---------------------------------------------------------------------------
// Kernel 2: WMMA pass. One wave computes one 16x16 tile of C.
//   dotV[m,t] = V_pred[m].V_gt[t]  (K=17 padded to 20 -> 5x wmma_f32_16x16x4_f32)
//   dotC[m,t] = C_pred[m].C_gt[t]  (K=2  padded to 4  -> 1x, pad half reads zeros)
//   out = class + 0.2*sqrt(max(nVp+nVg-2dotV,eps)) + 0.5*sqrt(max(nCp+nCg-2dotC,eps))
// Fragment layouts per ISA 7.12.2 (wave32):
//   A 16x4 f32: lane L holds M=L%16, K = 2*(L/16)+{0,1} in the 2 VGPRs
//   B 4x16 f32: lane L holds N=L%16, K = 2*(L/16)+{0,1}
//   D 16x16 f32: VGPR i, lane L -> M = i + 8*(L/16), N = L%16
// ---------------------------------------------------------------------------
__global__ __launch_bounds__(256) void cost_wmma_kernel(
    const float* __restrict__ padP, const float* __restrict__ padT,
    const int* __restrict__ ids,
    const float* __restrict__ negP, const float* __restrict__ nVp,
    const float* __restrict__ nCp, const float* __restrict__ nVg,
    const float* __restrict__ nCg, float* __restrict__ out) {
  const int lane = threadIdx.x & 31;
  const int wave = threadIdx.x >> 5;
  const int m16  = lane & 15;
  const int kh   = lane >> 4;                 // 0 or 1: K-half of the fragment
  const int n0   = (blockIdx.x * 8 + wave) * 16;
  const int t0   = blockIdx.y * 16;

  const float* Ap = padP + (n0 + m16) * PADW + 2 * kh;  // pred fragment row
  const float* Bp = padT + (t0 + m16) * PADW + 2 * kh;  // tgt  fragment row

  v8f accV = {};
  v8f accC = {};
#pragma unroll
  for (int kk = 0; kk < 5; ++kk) {            // K = 0..19 (17 valid, 3 zero-pad)
    v2f a = *(const v2f*)(Ap + kk * 4);
    v2f b = *(const v2f*)(Bp + kk * 4);
    accV = __builtin_amdgcn_wmma_f32_16x16x4_f32(
        false, a, false, b, (short)0, accV, false, false);
  }
  {                                           // K = 0..3 (C at pad[20..21], zeros above)
    v2f a = *(const v2f*)(Ap + 20);
    v2f b = *(const v2f*)(Bp + 20);
    accC = __builtin_amdgcn_wmma_f32_16x16x4_f32(
        false, a, false, b, (short)0, accC, false, false);
  }

  const int   t   = t0 + m16;
  const float nvg = nVg[t];
  const float ncg = nCg[t];
  const int   id  = ids[t];
#pragma unroll
  for (int i = 0; i < 8; ++i) {
    const int m = n0 + 8 * kh + i;            // D-matrix row held in VGPR i
    float ssV = nVp[m] + nvg - 2.0f * accV[i];
    float ssC = nCp[m] + ncg - 2.0f * accC[i];
    float viz = fast_sqrtf(fmaxf(ssV, 1e-12f));   // single v_sqrt_f32
    float ctr = fast_sqrtf(fmaxf(ssC, 1e-12f));
    out[m * NT + t] = negP[2 * m + id] + 0.2f * viz + 0.5f * ctr;
  }
}

// ---------------------------------------------------------------------------
// Kernel 3: L1 pass (VALU-bound core).
//   Each thread owns one target column t (tgt row cached in VGPRs), and loops
//   32 pred rows; pred addresses are block-uniform -> scalar (SMEM) loads that
//   co-issue with the VALU stream. Accumulates
//     sum_k |Vg[k/2]| * (0.5*|dZ_k| + 4*|dZ_k + dC_{k%2}|)
//   and adds into d_out (written by the WMMA pass).
// ---------------------------------------------------------------------------
__global__ __launch_bounds__(256) void cost_l1_kernel(
    const float* __restrict__ pkp, const float* __restrict__ tkp,
    float* __restrict__ out) {
  const int t = blockIdx.x * 256 + threadIdx.x;   // 0..511
  const float* Tr = tkp + t * KD;
  const float cg0 = Tr[0], cg1 = Tr[1];
  float zg[34], vg[17];
#pragma unroll
  for (int k = 0; k < 34; ++k) zg[k] = Tr[2 + k];
#pragma unroll
  for (int k = 0; k < 17; ++k) vg[k] = Tr[36 + k];

  const int nStart = blockIdx.y * 32;
  for (int n = nStart; n < nStart + 32; ++n) {
    const float* Pr = pkp + n * KD;               // uniform -> s_load
    if (n + 1 < NROWS)
      __builtin_prefetch((const void*)(Pr + KD), 0, 1);  // global_prefetch next row
    const float dc0 = Pr[0] - cg0;
    const float dc1 = Pr[1] - cg1;
    float acc = 0.f;
#pragma unroll
    for (int k = 0; k < 34; ++k) {
      float dz = Pr[2 + k] - zg[k];
      float da = dz + ((k & 1) ? dc1 : dc0);
      float u  = fmaf(fabsf(da), 4.0f, 0.5f * fabsf(dz));  // abs = free src modifier
      acc = fmaf(fabsf(vg[k >> 1]), u, acc);
    }
    out[n * NT + t] += acc;
  }
}

// ---------------------------------------------------------------------------
extern "C" void kernel_launch(void* const* d_in, const int* in_sizes, int n_in,
                              void* d_out, int out_size, void* d_ws, size_t ws_size,
                              hipStream_t stream) {
  const float* logits = (const float*)d_in[0];   // (16,512,2)
  const float* pkp    = (const float*)d_in[1];   // (16,512,53)
  const float* tkp    = (const float*)d_in[2];   // (512,53)
  const int*   ids    = (const int*)d_in[3];     // (512,)
  float* out = (float*)d_out;                    // (16,512,512)

  float* ws   = (float*)d_ws;
  float* negP = ws;                              // 16384 floats
  float* nVp  = ws + 16384;                      // 8192
  float* nCp  = ws + 24576;                      // 8192
  float* nVg  = ws + 32768;                      // 512
  float* nCg  = ws + 33280;                      // 512
  float* padP = ws + 33792;                      // 8192*24 = 196608
  float* padT = ws + 230400;                     // 512*24  = 12288  (total ~971 KB)

  prep_kernel<<<(NROWS + NT + 255) / 256, 256, 0, stream>>>(
      logits, pkp, tkp, negP, nVp, nCp, nVg, nCg, padP, padT);

  dim3 g1(NROWS / 128, NT / 16);                 // 64 x 32 blocks, 8 waves/block
  cost_wmma_kernel<<<g1, 256, 0, stream>>>(
      padP, padT, ids, negP, nVp, nCp, nVg, nCg, out);

  dim3 g2(NT / 256, NROWS / 32);                 // 2 x 256 blocks
  cost_l1_kernel<<<g2, 256, 0, stream>>>(pkp, tkp, out);
}